// enlargesegTextCNN2_77180562309385
// MI455X (gfx1250) — compile-verified
//
#include <hip/hip_runtime.h>
#include <hip/hip_bf16.h>
#include <math.h>

typedef __attribute__((ext_vector_type(2))) float v2f;
typedef __attribute__((ext_vector_type(8))) float v8f;

#define NB    4096
#define LL    2048
#define NKK   256
#define NCL   100
#define NCOL  3072   // 256*(3+4+5)
#define NROW  128    // C*16
// f-vector row offsets: K=3 -> [0,1536), K=4 -> [1536,2816), K=5 -> [2816,3840)

// ---------------------------------------------------------------------------
// Kernel A: gtab[ce][l] = be[l] + sum_d LUT[ce,d] * We[d,l]   (128 x 2048)
// ---------------------------------------------------------------------------
__global__ void k_gtab(const float* __restrict__ LUT, const float* __restrict__ We,
                       const float* __restrict__ be, float* __restrict__ gtab) {
  int ce = blockIdx.x;
  float l0 = LUT[ce * 4 + 0], l1 = LUT[ce * 4 + 1];
  float l2 = LUT[ce * 4 + 2], l3 = LUT[ce * 4 + 3];
  for (int l = threadIdx.x; l < LL; l += blockDim.x) {
    gtab[ce * LL + l] = be[l] + l0 * We[0 * LL + l] + l1 * We[1 * LL + l]
                              + l2 * We[2 * LL + l] + l3 * We[3 * LL + l];
  }
}

// ---------------------------------------------------------------------------
// Kernel B: Ptab = relu( gtab(128x2048) @ Bmat(2048x3072) )
// Bmat column (col = n*K + k per region) is just w{3,4,5} + col*2048.
// One wave (32 threads) per 16x16 tile; K-loop of v_wmma_f32_16x16x4_f32.
// ---------------------------------------------------------------------------
__global__ void __launch_bounds__(32)
k_ptab_wmma(const float* __restrict__ gtab,
            const float* __restrict__ w3, const float* __restrict__ w4,
            const float* __restrict__ w5, float* __restrict__ Ptab) {
  int nTile = blockIdx.x;      // 0..191
  int mTile = blockIdx.y;      // 0..7
  int lane  = threadIdx.x;     // 0..31
  int lo    = lane & 15;
  int half  = lane >> 4;       // 0: K pair {0,1}, 1: K pair {2,3}

  // A fragment source: row m = mTile*16 + lo
  const float* Arow = gtab + (mTile * 16 + lo) * LL;

  // B fragment source: column nn = nTile*16 + lo (region select is wave-uniform:
  // 768 and 1792 are multiples of 16)
  int nn = nTile * 16 + lo;
  const float* Bcol;
  if (nn < 768)        Bcol = w3 + nn * LL;
  else if (nn < 1792)  Bcol = w4 + (nn - 768) * LL;
  else                 Bcol = w5 + (nn - 1792) * LL;

  v8f acc = {};
  #pragma unroll 4
  for (int k0 = 0; k0 < LL; k0 += 4) {
    v2f a, b;
    a[0] = Arow[k0 + 2 * half];  a[1] = Arow[k0 + 2 * half + 1];
    b[0] = Bcol[k0 + 2 * half];  b[1] = Bcol[k0 + 2 * half + 1];
    acc = __builtin_amdgcn_wmma_f32_16x16x4_f32(
        /*neg_a=*/false, a, /*neg_b=*/false, b,
        /*c_mod=*/(short)0, acc, /*reuse_a=*/false, /*reuse_b=*/false);
  }

  // D layout: VGPR r -> M = mTile*16 + half*8 + r ; N = nTile*16 + lo
  #pragma unroll
  for (int r = 0; r < 8; ++r) {
    int mm = mTile * 16 + half * 8 + r;
    Ptab[mm * NCOL + nn] = fmaxf(acc[r], 0.0f);
  }
}

// ---------------------------------------------------------------------------
// Kernel C: Q[ce][j] = sum over (K,n,k) with p=c-k valid of
//                      Ptab[ce, col(K,n,k)] * W2[off_K + n*P_K + p, j]
// ---------------------------------------------------------------------------
__global__ void k_q(const float* __restrict__ Ptab, const float* __restrict__ W2,
                    float* __restrict__ Q) {
  int ce = blockIdx.x;          // 0..127
  int c  = ce >> 4;             // channel 0..7
  int j  = threadIdx.x;         // 128 threads, first 100 active
  if (j >= NCL) return;
  const float* Prow = Ptab + ce * NCOL;
  float acc = 0.0f;
  int col = 0;
  for (int n = 0; n < NKK; ++n)               // K=3, P=6, off=0
    for (int k = 0; k < 3; ++k, ++col) {
      int p = c - k;
      if (p >= 0 && p < 6) acc += Prow[col] * W2[(n * 6 + p) * NCL + j];
    }
  for (int n = 0; n < NKK; ++n)               // K=4, P=5, off=1536
    for (int k = 0; k < 4; ++k, ++col) {
      int p = c - k;
      if (p >= 0 && p < 5) acc += Prow[col] * W2[(1536 + n * 5 + p) * NCL + j];
    }
  for (int n = 0; n < NKK; ++n)               // K=5, P=4, off=2816
    for (int k = 0; k < 5; ++k, ++col) {
      int p = c - k;
      if (p >= 0 && p < 4) acc += Prow[col] * W2[(2816 + n * 4 + p) * NCL + j];
    }
  Q[ce * NCL + j] = acc;
}

// ---------------------------------------------------------------------------
// Kernel D: per-batch index computation + Q gather + log_softmax
// One 128-thread block per batch element.
// ---------------------------------------------------------------------------
__global__ void k_final(const int* __restrict__ x,
                        const float* __restrict__ len_emb, const float* __restrict__ ipd_emb,
                        const float* __restrict__ W1, const float* __restrict__ b1,
                        const float* __restrict__ S, const float* __restrict__ H,
                        const float* __restrict__ T, const float* __restrict__ Q,
                        const float* __restrict__ b2, float* __restrict__ out) {
  __shared__ int   idxs[8];
  __shared__ float red[128];
  int b = blockIdx.x;
  int t = threadIdx.x;

  if (t < 8) {
    int c = t;
    int xi0 = x[(b * 8 + c) * 2 + 0];
    int xi1 = x[(b * 8 + c) * 2 + 1];
    xi0 = xi0 < 0 ? 0 : (xi0 > 1499 ? 1499 : xi0);
    xi1 = xi1 < 0 ? 0 : (xi1 > 255 ? 255 : xi1);
    const float* le = len_emb + xi0 * 16;
    const float* ie = ipd_emb + xi1 * 16;

    float h[4];
    #pragma unroll
    for (int d = 0; d < 4; ++d) {
      float s = b1[d];
      #pragma unroll
      for (int i = 0; i < 16; ++i) s += le[i] * W1[i * 4 + d];
      #pragma unroll
      for (int i = 0; i < 16; ++i) s += ie[i] * W1[(16 + i) * 4 + d];
      h[d] = s;
    }
    float sgn[15];
    #pragma unroll
    for (int kk = 0; kk < 15; ++kk) {
      float m = -T[c * 15 + kk] - 1e-4f;
      #pragma unroll
      for (int d = 0; d < 4; ++d) m += h[d] * S[(c * 4 + d) * 15 + kk];
      sgn[kk] = (m > 0.0f) ? 1.0f : ((m < 0.0f) ? -1.0f : 0.0f);
    }
    int best = 0;
    float bestv = -3.4e38f;
    for (int j = 0; j < 16; ++j) {
      float v = 0.0f;
      #pragma unroll
      for (int kk = 0; kk < 15; ++kk) v += sgn[kk] * H[kk * 16 + j];
      if (v > bestv) { bestv = v; best = j; }   // strict >: first-max like argmax
    }
    idxs[c] = best;
  }
  __syncthreads();

  float logit = 0.0f;
  if (t < NCL) {
    logit = b2[t];
    #pragma unroll
    for (int c = 0; c < 8; ++c) logit += Q[(c * 16 + idxs[c]) * NCL + t];
  }

  // max reduce
  red[t] = (t < NCL) ? logit : -3.4e38f;
  __syncthreads();
  for (int s = 64; s > 0; s >>= 1) {
    if (t < s) red[t] = fmaxf(red[t], red[t + s]);
    __syncthreads();
  }
  float mx = red[0];
  __syncthreads();

  // sum-exp reduce
  red[t] = (t < NCL) ? __expf(logit - mx) : 0.0f;
  __syncthreads();
  for (int s = 64; s > 0; s >>= 1) {
    if (t < s) red[t] += red[t + s];
    __syncthreads();
  }
  float lse = __logf(red[0]);

  if (t < NCL) out[b * NCL + t] = (logit - mx) - lse;
}

// ---------------------------------------------------------------------------
extern "C" void kernel_launch(void* const* d_in, const int* in_sizes, int n_in,
                              void* d_out, int out_size, void* d_ws, size_t ws_size,
                              hipStream_t stream) {
  (void)in_sizes; (void)n_in; (void)out_size; (void)ws_size;
  const int*   x       = (const int*)  d_in[0];
  const float* len_emb = (const float*)d_in[1];
  const float* ipd_emb = (const float*)d_in[2];
  const float* W1      = (const float*)d_in[3];
  const float* b1      = (const float*)d_in[4];
  const float* We      = (const float*)d_in[5];
  const float* be      = (const float*)d_in[6];
  const float* w3      = (const float*)d_in[7];
  const float* w4      = (const float*)d_in[8];
  const float* w5      = (const float*)d_in[9];
  const float* W2      = (const float*)d_in[10];
  const float* b2      = (const float*)d_in[11];
  const float* S       = (const float*)d_in[12];
  const float* H       = (const float*)d_in[13];
  const float* T       = (const float*)d_in[14];
  const float* LUT     = (const float*)d_in[15];
  float* out = (float*)d_out;

  // workspace: gtab 128*2048 | Ptab 128*3072 | Q 128*100  (~2.7 MB total)
  float* gtab = (float*)d_ws;
  float* Ptab = gtab + NROW * LL;
  float* Q    = Ptab + NROW * NCOL;

  k_gtab<<<NROW, 256, 0, stream>>>(LUT, We, be, gtab);

  dim3 gB(NCOL / 16, NROW / 16);   // 192 x 8 tiles, one wave32 each
  k_ptab_wmma<<<gB, 32, 0, stream>>>(gtab, w3, w4, w5, Ptab);

  k_q<<<NROW, 128, 0, stream>>>(Ptab, W2, Q);

  k_final<<<NB, 128, 0, stream>>>(x, len_emb, ipd_emb, W1, b1, S, H, T, Q, b2, out);
}